// BroadcastObstaclesToLanes_25675314495799
// MI455X (gfx1250) — compile-verified
//
#include <hip/hip_runtime.h>
#include <hip/hip_bf16.h>

// Pure gather: out[i] = table[idx[i]], 16.7M int64 indices into a 1M x float2
// table. Roofline: ~128MB idx in + ~128MB out + 8MB table (L2-resident) at
// 23.3 TB/s => ~11us. Strategy:
//  - NT hints on the two touch-once streams (idx load, out store) so the
//    8MB gather table stays resident in the 192MB L2 (gathers use RT hints).
//  - 16B-per-lane transactions (b128) on both streams, wave32-coalesced.
//  - Grid-stride loop + global_prefetch_b8 of next iteration's index line.

typedef __attribute__((ext_vector_type(2))) long long v2i64;
typedef __attribute__((ext_vector_type(2))) float     v2f;
typedef __attribute__((ext_vector_type(4))) float     v4f;

__global__ __launch_bounds__(256) void
BroadcastObstaclesToLanes_gather_kernel(const v2f* __restrict__ table,
                                        const v2i64* __restrict__ idx2,
                                        v4f* __restrict__ out4,
                                        int npairs) // # of v4f outputs = M/2
{
  const int stride = gridDim.x * blockDim.x;
  for (int p = blockIdx.x * blockDim.x + threadIdx.x; p < npairs; p += stride) {
    // Prefetch next iteration's index cacheline (speculative, OOB is dropped).
    __builtin_prefetch(&idx2[p + stride], /*rw=*/0, /*locality=*/1);

    // Two int64 indices per lane as one 16B non-temporal load (read-once).
    v2i64 id = __builtin_nontemporal_load(&idx2[p]);

    // Random 8B gathers; default RT hint keeps the 8MB table hot in L2.
    v2f a = table[(int)id.x];
    v2f b = table[(int)id.y];

    v4f o;
    o.x = a.x; o.y = a.y;
    o.z = b.x; o.w = b.y;

    // 16B non-temporal store (write-once stream, don't pollute L2).
    __builtin_nontemporal_store(o, &out4[p]);
  }
}

extern "C" void kernel_launch(void* const* d_in, const int* in_sizes, int n_in,
                              void* d_out, int out_size, void* d_ws, size_t ws_size,
                              hipStream_t stream) {
  const v2f*   table = (const v2f*)d_in[0];    // obs_pos: float32 (N_OBS, 2)
  const v2i64* idx2  = (const v2i64*)d_in[1];  // same_obs_mask: int64 (M, 1), as pairs
  v4f*         out4  = (v4f*)d_out;            // float32 (M, 2), as float4 pairs

  const int npairs  = out_size / 4;            // out_size = M*2 floats; 4 per v4f
  const int threads = 256;
  int blocks = 8192;                           // grid-stride: ~4 iters/thread at M=16.7M
  const int need = (npairs + threads - 1) / threads;
  if (blocks > need) blocks = need;
  if (blocks < 1)    blocks = 1;

  BroadcastObstaclesToLanes_gather_kernel<<<blocks, threads, 0, stream>>>(
      table, idx2, out4, npairs);
}